// GaugePersonalizedLayer_46265387712744
// MI455X (gfx1250) — compile-verified
//
#include <hip/hip_runtime.h>

// GaugePersonalizedLayer for MI455X (gfx1250, wave32, WMMA).
//
// Fusion:  h_out = h_t + h_t @ Wmod_b^T @ W_up^T
//   with   Wmod_b = alpha * (Q_b - I) @ W_down,  Q_b = cayley(A_u[b])
//
// Memory-bound problem (~0.5 GB min traffic -> ~23 us @ 23.3 TB/s), so we do
// one pass over h_t: each workgroup stages a 16x4096 fp32 tile in LDS (256 KB,
// only possible on CDNA5's 320 KB WGP LDS), then runs both GEMMs with
// V_WMMA_F32_16X16X4_F32 to keep full fp32 precision. The tile staging uses
// GLOBAL_LOAD_ASYNC_TO_LDS_B128 (ASYNCcnt-tracked DMA, no VGPR round-trip)
// when the toolchain exposes the builtin.

typedef __attribute__((ext_vector_type(2))) float v2f;
typedef __attribute__((ext_vector_type(4))) float v4f;
typedef __attribute__((ext_vector_type(8))) float v8f;
typedef __attribute__((ext_vector_type(4))) int   v4i;

typedef __attribute__((address_space(1))) v4i* gv4i_p;  // global int4*
typedef __attribute__((address_space(3))) v4i* lv4i_p;  // LDS int4*

#define D_MODEL 4096
#define D_FIBER 64
#define B_SZ    4
#define L_SZ    4096
#define TILE_M  16
#define H_STRIDE 4100   // 4096 + 4 pad: lanes hit distinct LDS banks across rows
#define DELTA_STRIDE 68 // 64 + 4 pad

// ---------------------------------------------------------------------------
// Kernel 1: Cayley transform per batch via Gauss-Jordan on (I-A) Q = (I+A).
// One block per batch, 64 threads; thread t owns column t. Emits C = Q - I.
// ---------------------------------------------------------------------------
__global__ void cayley_kernel(const float* __restrict__ A_u,
                              float* __restrict__ Cmat) {
  __shared__ float M[64][65];
  __shared__ float R[64][65];
  __shared__ float fac[64];
  const int b = blockIdx.x;
  const int t = threadIdx.x;
  const float* A = A_u + b * 64 * 64;

  for (int i = 0; i < 64; ++i) {
    float a  = A[i * 64 + t];
    float id = (i == t) ? 1.0f : 0.0f;
    M[i][t] = id - a;   // I - A
    R[i][t] = id + a;   // I + A
  }
  __syncthreads();

  for (int k = 0; k < 64; ++k) {
    float piv = M[k][k];          // read pivot before anyone scales row k
    __syncthreads();
    float invp = 1.0f / piv;      // (I - skew) is always nonsingular
    M[k][t] *= invp;
    R[k][t] *= invp;
    __syncthreads();
    float mkt = M[k][t];
    float rkt = R[k][t];
    fac[t] = (t == k) ? 0.0f : M[t][k];   // snapshot column k factors
    __syncthreads();
    for (int i = 0; i < 64; ++i) {
      float f = fac[i];
      M[i][t] -= f * mkt;
      R[i][t] -= f * rkt;
    }
    __syncthreads();
  }

  // C = Q - I  (R now holds Q)
  for (int g = 0; g < 64; ++g)
    Cmat[(b * 64 + g) * 64 + t] = R[g][t] - ((g == t) ? 1.0f : 0.0f);
}

// ---------------------------------------------------------------------------
// Kernel 2: Wmod[b][g][d] = alpha * sum_f C[b][g][f] * W_down[f][d]
// 4*64*4096 outputs, one per thread, 64-MAC loop. Negligible cost (67 M MAC).
// ---------------------------------------------------------------------------
__global__ void wmod_kernel(const float* __restrict__ Cmat,
                            const float* __restrict__ W_down,
                            const float* __restrict__ alpha_p,
                            float* __restrict__ Wmod) {
  const int idx = blockIdx.x * blockDim.x + threadIdx.x; // B*64*4096 total
  const int b = idx >> 18;          // 64*4096 = 2^18
  const int r = idx & 262143;
  const int g = r >> 12;            // 4096 = 2^12
  const int d = r & 4095;
  const float* C = Cmat + (b * 64 + g) * 64;
  float acc = 0.0f;
#pragma unroll 8
  for (int f = 0; f < 64; ++f) acc += C[f] * W_down[f * D_MODEL + d];
  Wmod[idx] = acc * (*alpha_p);
}

// ---------------------------------------------------------------------------
// Kernel 3: main fused pass. grid = B * (L/16) workgroups, 128 threads (4
// waves). Stage 2: delta(16x64) = h_tile @ Wmod_b^T (K=4096). Stage 3:
// h_out = h_tile + delta @ W_up^T (K=64). Both via wmma_f32_16x16x4_f32.
//
// WMMA fp32 lane layout (ISA 7.12.2):
//   A 16x4 : lane m=lane&15 holds row m; lanes 0-15 carry K={k,k+1},
//            lanes 16-31 carry K={k+2,k+3} in VGPR0/VGPR1.
//   B 4x16 : lane n=lane&15 holds col n; same K split as A.
//   C/D    : VGPR v -> rows v (lanes 0-15) and v+8 (lanes 16-31), col lane&15.
// ---------------------------------------------------------------------------
__global__ void gauge_main_kernel(const float* __restrict__ h_t,
                                  const float* __restrict__ Wmod,
                                  const float* __restrict__ W_up,
                                  float* __restrict__ h_out) {
  extern __shared__ float smem[];
  float* lds_h     = smem;                          // TILE_M * H_STRIDE
  float* lds_delta = smem + TILE_M * H_STRIDE;      // TILE_M * DELTA_STRIDE

  const int tid  = threadIdx.x;     // 0..127
  const int wave = tid >> 5;        // 0..3
  const int lane = tid & 31;
  const int b    = blockIdx.x >> 8;         // L/16 = 256 tiles per batch
  const int l0   = (blockIdx.x & 255) * TILE_M;

  const size_t tile_base = (size_t)(b * L_SZ + l0) * D_MODEL;
  const float* hsrc = h_t + tile_base;

  // ---- Stage 1: stage 16x4096 fp32 tile into LDS (row stride padded) ----
#if __has_builtin(__builtin_amdgcn_global_load_async_to_lds_b128)
  {
    // CDNA5 async DMA path: 16 B per lane straight into LDS, ASYNCcnt-tracked.
    for (int i = tid; i < (TILE_M * D_MODEL) / 4; i += 128) {
      const int r = i >> 10;        // row (1024 float4 per row)
      const int c = i & 1023;
      float* ldst = lds_h + r * H_STRIDE + c * 4;      // 16 B aligned
      const float* gsrc = hsrc + i * 4;                // 16 B aligned
      __builtin_amdgcn_global_load_async_to_lds_b128(
          (gv4i_p)gsrc, (lv4i_p)ldst, /*imm offset*/ 0, /*cpol*/ 0);
    }
#if __has_builtin(__builtin_amdgcn_s_wait_asynccnt)
    __builtin_amdgcn_s_wait_asynccnt(0);
#else
    asm volatile("s_wait_asynccnt 0x0" ::: "memory");
#endif
  }
#else
  {
    // Fallback: VGPR-bounce copy (global_load_b128 + ds_store_b128).
    const v4f* src4 = (const v4f*)hsrc;               // 16384 x float4, contiguous
    for (int i = tid; i < (TILE_M * D_MODEL) / 4; i += 128) {
      const int r = i >> 10;
      const int c = i & 1023;
      v4f* dst4 = (v4f*)(lds_h + r * H_STRIDE);       // H_STRIDE%4==0 -> aligned
      dst4[c] = src4[i];
    }
  }
#endif
  __syncthreads();

  const int m  = lane & 15;
  const int hi = lane >> 4;         // 0: K lower pair, 1: K upper pair

  // ---- Stage 2: delta = h_tile @ Wmod_b^T ; wave w owns fiber cols w*16.. ----
  {
    const int g0 = wave * 16;
    const float* Arow = lds_h + m * H_STRIDE;
    const float* Brow = Wmod + (size_t)b * D_FIBER * D_MODEL
                             + (size_t)(g0 + m) * D_MODEL;  // B[d][n]=Wmod[n][d]
    v8f acc = {};
    for (int k = 0; k < D_MODEL; k += 4) {
      v2f a, bb;
      a.x  = Arow[k + 2 * hi];
      a.y  = Arow[k + 2 * hi + 1];
      bb.x = Brow[k + 2 * hi];
      bb.y = Brow[k + 2 * hi + 1];
      acc = __builtin_amdgcn_wmma_f32_16x16x4_f32(
          false, a, false, bb, (short)0, acc, false, false);
    }
#pragma unroll
    for (int v = 0; v < 8; ++v)
      lds_delta[(v + 8 * hi) * DELTA_STRIDE + g0 + m] = acc[v];
  }
  __syncthreads();

  // ---- Stage 3: h_out = h_tile + delta @ W_up^T, 256 N-tiles over 4 waves ----
  float* outp = h_out + tile_base;
  const float* Adelta = lds_delta + m * DELTA_STRIDE;
  for (int nt = wave; nt < D_MODEL / 16; nt += 4) {
    const int n0 = nt * 16;
    const float* Bcol = W_up + (size_t)(n0 + m) * D_FIBER; // B[g][n]=W_up[n][g]
    v8f acc = {};
#pragma unroll
    for (int k = 0; k < D_FIBER; k += 4) {
      v2f a, bb;
      a.x  = Adelta[k + 2 * hi];
      a.y  = Adelta[k + 2 * hi + 1];
      bb.x = Bcol[k + 2 * hi];
      bb.y = Bcol[k + 2 * hi + 1];
      acc = __builtin_amdgcn_wmma_f32_16x16x4_f32(
          false, a, false, bb, (short)0, acc, false, false);
    }
#pragma unroll
    for (int v = 0; v < 8; ++v) {
      const int mm = v + 8 * hi;
      outp[(size_t)mm * D_MODEL + n0 + m] =
          lds_h[mm * H_STRIDE + n0 + m] + acc[v];
    }
  }
}

// ---------------------------------------------------------------------------
// Launch. Inputs: [0]=h_t (B,L,4096) [1]=z0 (unused by reference) [2]=A_u
// (B,64,64) [3]=alpha scalar [4]=W_down (64,4096) [5]=W_up (4096,64).
// Workspace layout: [0, 64KB) Cmat (B*64*64 fp32); [64KB, 64KB+4MB) Wmod.
// ---------------------------------------------------------------------------
extern "C" void kernel_launch(void* const* d_in, const int* in_sizes, int n_in,
                              void* d_out, int out_size, void* d_ws, size_t ws_size,
                              hipStream_t stream) {
  (void)in_sizes; (void)n_in; (void)out_size; (void)ws_size;
  const float* h_t    = (const float*)d_in[0];
  const float* A_u    = (const float*)d_in[2];
  const float* alpha  = (const float*)d_in[3];
  const float* W_down = (const float*)d_in[4];
  const float* W_up   = (const float*)d_in[5];
  float* out = (float*)d_out;

  float* Cmat = (float*)d_ws;                              // 4*64*64 floats
  float* Wmod = (float*)((char*)d_ws + 64 * 1024);         // 4*64*4096 floats

  cayley_kernel<<<B_SZ, 64, 0, stream>>>(A_u, Cmat);

  wmod_kernel<<<(B_SZ * D_FIBER * D_MODEL) / 256, 256, 0, stream>>>(
      Cmat, W_down, alpha, Wmod);

  const size_t smem_bytes =
      (size_t)(TILE_M * H_STRIDE + TILE_M * DELTA_STRIDE) * sizeof(float);
  gauge_main_kernel<<<B_SZ * (L_SZ / TILE_M), 128, smem_bytes, stream>>>(
      h_t, Wmod, W_up, out);
}